// SpanMaskingStrategy_88261577933276
// MI455X (gfx1250) — compile-verified
//
#include <hip/hip_runtime.h>
#include <hip/hip_bf16.h>
#include <stdint.h>

// Span masking: one workgroup (4 wave32s) per row.
//  - wave 0 issues a TDM tensor_load_to_lds of the row's tokens (32 KB) at entry
//  - lane 0 runs the serial span-sampling loop on a 1 KiB LDS bitset, counting
//    newly masked tokens incrementally (popc of changed bits)
//  - TDM copy overlaps the RNG loop; s_wait_tensorcnt 0 before the output sweep
//  - all 128 threads then write x_masked (float) and mask (float 0/1), coalesced

#define SEQ        8192
#define MASK_WORDS (SEQ / 32)
#define MASK_ID    50256

#if defined(__has_builtin)
#if __has_builtin(__builtin_amdgcn_tensor_load_to_lds)
#define USE_TDM 1
#endif
#if __has_builtin(__builtin_amdgcn_s_wait_tensorcnt)
#define HAVE_TENSORCNT_WAIT 1
#endif
#endif

typedef uint32_t u32x4 __attribute__((ext_vector_type(4)));
typedef int      i32x8 __attribute__((ext_vector_type(8)));
typedef int      i32x4 __attribute__((ext_vector_type(4)));

__device__ __forceinline__ uint32_t pcg32(uint64_t& s) {
    uint64_t old = s;
    s = old * 6364136223846793005ull + 1442695040888963407ull;
    uint32_t xorshifted = (uint32_t)(((old >> 18u) ^ old) >> 27u);
    uint32_t rot = (uint32_t)(old >> 59u);
    return (xorshifted >> rot) | (xorshifted << ((32u - rot) & 31u));
}

__global__ __launch_bounds__(128) void span_mask_kernel(
    const int* __restrict__ x0, const float* __restrict__ mask_prob,
    float* __restrict__ out_x, float* __restrict__ out_m)
{
    __shared__ int      toks[SEQ];          // 32 KB staged row of tokens
    __shared__ uint32_t mbits[MASK_WORDS];  // 1 KB mask bitset

    const int row = blockIdx.x;
    const int tid = threadIdx.x;

#if defined(USE_TDM)
    if (tid < 32) {
        // Build Tensor DMA Descriptor (D#), groups 0 & 1; 1-row 2D tile of
        // 8192 x 4-byte elements: global row -> LDS toks[]. Issued once by wave 0.
        uint64_t ga = (uint64_t)(uintptr_t)(x0 + (size_t)row * SEQ);
        uint32_t la = (uint32_t)(uintptr_t)&toks[0];   // low 32 bits = LDS byte offset

        u32x4 g0;
        g0.x = 1u;                                       // count=1 (valid), user mode
        g0.y = la;                                       // lds_addr
        g0.z = (uint32_t)ga;                             // global_addr[31:0]
        g0.w = ((uint32_t)(ga >> 32) & 0x01FFFFFFu)      // global_addr[56:32]
             | (2u << 30);                               // type = 2 ("image")

        i32x8 g1;
        g1[0] = 0x20000;                                 // data_size=2 (4 B), wg_mask=0
        g1[1] = (int)((SEQ & 0xFFFF) << 16);             // tensor_dim0[15:0]
        g1[2] = (int)(((SEQ >> 16) & 0xFFFF)             // tensor_dim0[31:16]
              | (1 << 16));                              // tensor_dim1[15:0] = 1
        g1[3] = (int)((SEQ & 0xFFFF) << 16);             // tensor_dim1[31:16]=0 | tile_dim0=SEQ
        g1[4] = 1;                                       // tile_dim1=1, tile_dim2=0
        g1[5] = SEQ;                                     // tensor_dim0_stride[31:0]
        g1[6] = 0;                                       // stride[47:32]=0, dim1_stride lo=0
        g1[7] = 0;
        i32x4 z4; z4[0] = 0; z4[1] = 0; z4[2] = 0; z4[3] = 0;     // groups 2/3 unused (2D)
        i32x8 z8; z8[0] = 0; z8[1] = 0; z8[2] = 0; z8[3] = 0;
                  z8[4] = 0; z8[5] = 0; z8[6] = 0; z8[7] = 0;     // extra group (clang-23 form)

        // amdgpu-toolchain (clang-23) 6-arg form:
        // (uint32x4 g0, int32x8 g1, int32x4, int32x4, int32x8, i32 cpol)
        __builtin_amdgcn_tensor_load_to_lds(g0, g1, z4, z4, z8, 0);
    }
#else
    for (int i = tid; i < SEQ; i += blockDim.x)
        toks[i] = x0[(size_t)row * SEQ + i];
#endif

    for (int w = tid; w < MASK_WORDS; w += blockDim.x) mbits[w] = 0u;
    __syncthreads();

    // ---- serial span-sampling loop (overlaps the TDM copy) ----
    if (tid == 0) {
        const float p      = mask_prob[row];
        const int   target = (int)floorf(p * (float)SEQ);
        const float inv_log1mq = -2.4663035f;            // 1 / ln(1 - 1/3)
        uint64_t state = 0x9E3779B97F4A7C15ull * (uint64_t)(row + 1) ^ 0x2545F4914F6CDD1Dull;
        int cur = 0;
        while (cur < target) {
            uint32_t r1 = pcg32(state);
            uint32_t r2 = pcg32(state);
            int start = (int)(r1 & (SEQ - 1));
            float u = (float)(r2 >> 8) * (1.0f / 16777216.0f);
            u = fmaxf(u, 1e-7f);
            int geo = (int)floorf(__logf(u) * inv_log1mq) + 1;   // Geometric(1/3) + 1
            int len = min(geo, SEQ - start);
            int end = start + len;
            int w0 = start >> 5, w1 = (end - 1) >> 5;
            for (int w = w0; w <= w1; ++w) {
                int base = w << 5;
                int lo = max(start, base) - base;
                int hi = min(end, base + 32) - base;     // (lo, hi] in [1,32]
                uint32_t bits = ((hi == 32) ? 0xFFFFFFFFu : ((1u << hi) - 1u))
                              & ~((1u << lo) - 1u);
                uint32_t old = mbits[w];
                uint32_t nw  = old | bits;
                if (nw != old) { mbits[w] = nw; cur += __popc(nw ^ old); }
            }
        }
    }

#if defined(USE_TDM) && defined(HAVE_TENSORCNT_WAIT)
    if (tid < 32) __builtin_amdgcn_s_wait_tensorcnt(0); // wave 0: DMA complete
#endif
    __syncthreads();

    // ---- output sweep: all 4 waves, coalesced ----
    const size_t base = (size_t)row * SEQ;
    for (int i = tid; i < SEQ; i += blockDim.x) {
        bool b = (mbits[i >> 5] >> (i & 31)) & 1u;
        int tok = toks[i];
        out_x[base + i] = (float)(b ? MASK_ID : tok);
        out_m[base + i] = b ? 1.0f : 0.0f;
    }
}

extern "C" void kernel_launch(void* const* d_in, const int* in_sizes, int n_in,
                              void* d_out, int out_size, void* d_ws, size_t ws_size,
                              hipStream_t stream) {
    const int*   x0 = (const int*)d_in[0];
    const float* mp = (const float*)d_in[1];
    const int B = in_sizes[1];                 // mask_prob has one entry per row
    float* out_x = (float*)d_out;              // x_masked (as float), then mask
    float* out_m = out_x + (size_t)B * SEQ;
    span_mask_kernel<<<dim3(B), dim3(128), 0, stream>>>(x0, mp, out_x, out_m);
}